// RDWTFrontEnd_15109694947403
// MI455X (gfx1250) — compile-verified
//
#include <hip/hip_runtime.h>

// ---- types ----
typedef float v2f  __attribute__((ext_vector_type(2)));
typedef float v8f  __attribute__((ext_vector_type(8)));
typedef unsigned int u32x4 __attribute__((ext_vector_type(4)));
typedef int   i32x8 __attribute__((ext_vector_type(8)));
typedef int   i32x4 __attribute__((ext_vector_type(4)));

#define T_LEN   8192
#define KLEN    16
#define PAD_L   8
#define ROWBUF  (T_LEN + 16)      // 8 halo left + 8192 + 8 halo right = 8208
#define LEVELS  4

// ---------------- prep: compute per-level filters / tau / alpha ----------------
__device__ __forceinline__ float resample_tap(const float* p, float s, int t) {
    const float c = 7.5f;                      // 0.5*(K-1), K=16
    float u  = c + ((float)t - c) / s;
    float fu = floorf(u);
    float u0f = fminf(fmaxf(fu, 0.0f), 15.0f); // clip(floor(u),0,15)
    int   u0  = (int)u0f;
    int   u1  = (u0 + 1 > 15) ? 15 : u0 + 1;
    float w1  = u - u0f;
    return (1.0f - w1) * p[u0] + w1 * p[u1];
}

__global__ void rdwt_prep(const float* __restrict__ lo_proto,
                          const float* __restrict__ hi_proto,
                          const float* __restrict__ z,
                          const float* __restrict__ tau_raw,
                          const float* __restrict__ alpha,
                          float* __restrict__ ws) {
    int t = threadIdx.x;                 // 128 threads
    if (t < 128) {
        int lvl = t >> 5, which = (t >> 4) & 1, tap = t & 15;
        // s = 1 + (MAX_SCALE-1) * sigmoid(z / TEMP),  TEMP=0.5
        float s = 1.0f + 3.0f / (1.0f + __expf(-2.0f * z[lvl]));
        const float* p = which ? hi_proto : lo_proto;
        float sum = 0.0f;
        #pragma unroll
        for (int j = 0; j < 16; ++j) sum += fabsf(resample_tap(p, s, j));
        ws[t] = resample_tap(p, s, tap) / (sum + 1e-6f);
    }
    if (t < LEVELS) {
        float xr = tau_raw[t];
        ws[128 + t] = (xr > 20.0f) ? xr : log1pf(__expf(xr));  // softplus
        ws[132 + t] = alpha[t];
    }
}

// ---------------- main fused cascade ----------------
__device__ __forceinline__ float toep(const float* f, int k, int n) {
    int j = k - n;                       // B[k][n] = f[k-n] on the band
    return (j >= 0 && j < KLEN) ? f[j] : 0.0f;
}

__global__ __launch_bounds__(256)
void rdwt_main(const float* __restrict__ x,
               const float* __restrict__ fw,
               float* __restrict__ out) {
    __shared__ __align__(16) float abuf[2][ROWBUF];
    __shared__ float filt[136];          // 128 taps + 4 tau + 4 alpha

    const int tid  = threadIdx.x;
    const int wave = tid >> 5;
    const int lane = tid & 31;
    const int mrow = lane & 15;          // A-matrix M row for this lane
    const int khalf = (lane >> 4) << 1;  // lanes 16..31 hold K rows +2 (A & B frags)
    const int chalf = (lane >> 4) << 3;  // lanes 16..31 hold C rows M+8
    const int ncol  = lane & 15;         // B/C column

    const float* xrow = x   + (size_t)blockIdx.x * T_LEN;
    float*       orow = out + (size_t)blockIdx.x * T_LEN;

    if (tid < 136) filt[tid] = fw[tid];
    if (tid < 8) {                       // zero halos of both buffers once
        abuf[0][tid] = 0.0f; abuf[0][T_LEN + 8 + tid] = 0.0f;
        abuf[1][tid] = 0.0f; abuf[1][T_LEN + 8 + tid] = 0.0f;
    }

#if __has_builtin(__builtin_amdgcn_tensor_load_to_lds) && __has_builtin(__builtin_amdgcn_s_wait_tensorcnt)
    // Tensor Data Mover: DMA the whole row (8192 fp32) into LDS interior.
    if (tid < 32) {                      // issue once (wave 0)
        unsigned long long ga = (unsigned long long)(size_t)xrow;
        unsigned lds_off = (unsigned)(size_t)(&abuf[0][PAD_L]);
        u32x4 g0;
        g0[0] = 1u;                                       // count=1, user mode
        g0[1] = lds_off;                                  // lds_addr
        g0[2] = (unsigned)(ga & 0xFFFFFFFFu);             // global_addr[31:0]
        g0[3] = (unsigned)((ga >> 32) & 0x01FFFFFFu)      // global_addr[56:32]
              | (2u << 30);                               // type=2 ("image")
        i32x8 g1;
        g1[0] = 0x00020000;   // workgroup_mask=0, data_size=2 (4 bytes)
        g1[1] = 0x20000000;   // tensor_dim0[15:0]=8192 in bits 31:16
        g1[2] = 0x00010000;   // tensor_dim0[31:16]=0, tensor_dim1[15:0]=1
        g1[3] = 0x20000000;   // tensor_dim1[31:16]=0, tile_dim0=8192
        g1[4] = 0x00000001;   // tile_dim1=1, tile_dim2=0
        g1[5] = 0x00002000;   // tensor_dim0_stride=8192
        g1[6] = 0;
        g1[7] = 0;
        i32x4 gz4 = {0, 0, 0, 0};
        i32x8 gz8 = {0, 0, 0, 0, 0, 0, 0, 0};
        __builtin_amdgcn_tensor_load_to_lds(g0, g1, gz4, gz4, gz8, 0);
        __builtin_amdgcn_s_wait_tensorcnt(0);
    }
#else
    for (int i = tid; i < T_LEN / 4; i += 256)
        *(float4*)&abuf[0][PAD_L + 4 * i] = ((const float4*)xrow)[i];
#endif
    __syncthreads();

    v8f dsum[4] = {};                    // per-wave detail accumulators (4 tiles)
    int cur = 0;

    #pragma unroll 1
    for (int lvl = 0; lvl < LEVELS; ++lvl) {
        const float* flo = &filt[lvl * 32];
        const float* fhi = flo + 16;
        const float tau   = filt[128 + lvl];
        const float alpha = filt[132 + lvl];

        // Build banded-Toeplitz B fragments (32x16) for low & high filters.
        v2f Blo[8], Bhi[8];
        #pragma unroll
        for (int q = 0; q < 8; ++q) {
            int k0 = 4 * q + khalf;
            Blo[q].x = toep(flo, k0,     ncol);
            Blo[q].y = toep(flo, k0 + 1, ncol);
            Bhi[q].x = toep(fhi, k0,     ncol);
            Bhi[q].y = toep(fhi, k0 + 1, ncol);
        }

        const float* ab = &abuf[cur][0];
        float*       nb = &abuf[cur ^ 1][0];

        #pragma unroll
        for (int tile = 0; tile < 4; ++tile) {
            const int t0 = ((wave << 2) + tile) << 8;   // 256 outputs per tile
            const int abase = t0 + 16 * mrow + khalf;

            v2f Af[8];
            #pragma unroll
            for (int q = 0; q < 8; ++q)
                Af[q] = *(const v2f*)&ab[abase + 4 * q];

            v8f alo = {}, ahi = {};
            #pragma unroll
            for (int q = 0; q < 8; ++q)
                alo = __builtin_amdgcn_wmma_f32_16x16x4_f32(
                        false, Af[q], false, Blo[q], (short)0, alo, false, false);
            #pragma unroll
            for (int q = 0; q < 8; ++q)
                ahi = __builtin_amdgcn_wmma_f32_16x16x4_f32(
                        false, Af[q], false, Bhi[q], (short)0, ahi, false, false);

            const int obase = PAD_L + t0 + (chalf << 4) + ncol;
            #pragma unroll
            for (int r = 0; r < 8; ++r) {
                float d  = ahi[r];
                float th = fabsf(d) - tau;
                float sd = (th > 0.0f) ? ((d > 0.0f) ? th : -th) : 0.0f;
                dsum[tile][r] += alpha * sd;
                nb[obase + 16 * r] = alo[r];             // low-pass to next level
            }
        }
        __syncthreads();
        cur ^= 1;
    }

    // final = low-pass + detail sum; stage in spare buffer for coalesced stores
    {
        const float* af    = &abuf[cur][0];
        float*       stage = &abuf[cur ^ 1][0];
        #pragma unroll
        for (int tile = 0; tile < 4; ++tile) {
            const int t0   = ((wave << 2) + tile) << 8;
            const int base = t0 + (chalf << 4) + ncol;
            #pragma unroll
            for (int r = 0; r < 8; ++r)
                stage[base + 16 * r] = af[PAD_L + base + 16 * r] + dsum[tile][r];
        }
    }
    __syncthreads();
    {
        const float* stage = &abuf[cur ^ 1][0];
        for (int i = tid; i < T_LEN / 4; i += 256)
            ((float4*)orow)[i] = *(const float4*)&stage[4 * i];
    }
}

// ---------------- host launcher ----------------
extern "C" void kernel_launch(void* const* d_in, const int* in_sizes, int n_in,
                              void* d_out, int out_size, void* d_ws, size_t ws_size,
                              hipStream_t stream) {
    (void)in_sizes; (void)n_in; (void)out_size; (void)ws_size;
    const float* x        = (const float*)d_in[0];
    const float* lo_proto = (const float*)d_in[1];
    const float* hi_proto = (const float*)d_in[2];
    const float* z        = (const float*)d_in[3];
    const float* tau_raw  = (const float*)d_in[4];
    const float* alpha    = (const float*)d_in[5];
    float* out = (float*)d_out;
    float* ws  = (float*)d_ws;

    rdwt_prep<<<1, 128, 0, stream>>>(lo_proto, hi_proto, z, tau_raw, alpha, ws);
    rdwt_main<<<32 * 128, 256, 0, stream>>>(x, ws, out);
}